// NTFNet_25898652795550
// MI455X (gfx1250) — compile-verified
//
#include <hip/hip_runtime.h>
#include <stdint.h>

// NTFNet multiplicative update, fused single pass.
// Shifts are (0,-8) and (0,+8): every (channel,row) is independent -> one
// workgroup per row, whole row pipeline (front->middle->rear) in LDS.
// Memory-bound: ~200 MB total traffic -> ~8.6us at 23.3 TB/s.
// Uses CDNA5 async global<->LDS DMA path (ASYNCcnt) for row staging.

#define Hh   1080
#define Ww   1920
#define SHFT 8
#define EPS  1e-5f

__device__ __forceinline__ uint32_t lds_off(const void* p) {
  // shared-var generic pointer: low 32 bits are the LDS byte offset
  return (uint32_t)(uintptr_t)p;
}

__device__ __forceinline__ void async_ld_b128(uint32_t lds, const float* g) {
  asm volatile("global_load_async_to_lds_b128 %0, %1, off"
               :: "v"(lds), "v"(g) : "memory");
}
__device__ __forceinline__ void async_st_b128(float* g, uint32_t lds) {
  asm volatile("global_store_async_from_lds_b128 %0, %1, off"
               :: "v"(g), "v"(lds) : "memory");
}
__device__ __forceinline__ void wait_async0() {
  asm volatile("s_wait_asynccnt 0" ::: "memory");
}

__device__ __forceinline__ int wrapW(int x) {
  if (x < 0) x += Ww;
  else if (x >= Ww) x -= Ww;
  return x;
}
// mask(0, s): s>=0 zeros columns [0,s); s<0 zeros columns [W+s, W)
__device__ __forceinline__ float maskv(int x, int s) {
  return (s >= 0) ? (x >= s ? 1.0f : 0.0f) : (x < Ww + s ? 1.0f : 0.0f);
}
__device__ __forceinline__ float clamp01(float v) {
  return fminf(fmaxf(v, 0.0f), 1.0f);
}

__global__ __launch_bounds__(256) void ntf_row_kernel(
    const float* __restrict__ front, const float* __restrict__ middle,
    const float* __restrict__ rear, const float* __restrict__ images,
    float* __restrict__ out) {
  __shared__ __align__(16) float sF[Ww];   // old front, later rear'
  __shared__ __align__(16) float sM[Ww];   // middle, updated in place
  __shared__ __align__(16) float sR[Ww];   // old rear
  __shared__ __align__(16) float sI0[Ww];
  __shared__ __align__(16) float sI1[Ww];
  __shared__ __align__(16) float sFN[Ww];  // new front

  const int tid = threadIdx.x;
  const int b = blockIdx.x;
  const int c = b / Hh;
  const int h = b - c * Hh;
  const long plane  = (long)Hh * (long)Ww;       // H*W
  const long rowoff = ((long)c * Hh + h) * (long)Ww;

  const float* gF  = front  + rowoff;
  const float* gM  = middle + rowoff;
  const float* gR  = rear   + rowoff;
  const float* gI0 = images + ((long)(0 * 3 + c) * Hh + h) * (long)Ww;
  const float* gI1 = images + ((long)(1 * 3 + c) * Hh + h) * (long)Ww;

  // ---- stage 0: async DMA the five rows into LDS (b128 per lane) ----
  for (int q = tid; q < Ww / 4; q += 256) {
    const int x = q * 4;
    async_ld_b128(lds_off(&sF[x]),  gF  + x);
    async_ld_b128(lds_off(&sM[x]),  gM  + x);
    async_ld_b128(lds_off(&sR[x]),  gR  + x);
    async_ld_b128(lds_off(&sI0[x]), gI0 + x);
    async_ld_b128(lds_off(&sI1[x]), gI1 + x);
  }
  wait_async0();
  __syncthreads();

  const int SW[2] = {-SHFT, SHFT};  // SHIFTS order: (0,-8) then (0,+8)

  // ---- stage 1: front' = clip(F*(num+e)/(den+e))  ->  sFN ----
  for (int x = tid; x < Ww; x += 256) {
    float num = 0.f, den = 0.f;
#pragma unroll
    for (int i = 0; i < 2; ++i) {
      const int sw = SW[i];
      const int y = wrapW(x + sw);              // roll by anti = -sw
      const float ma = maskv(x, -sw);
      const float I = (i == 0) ? sI0[y] : sI1[y];
      // add evaluated at y with OLD state
      const float add = sF[wrapW(y - sw)] * maskv(y, sw) + sM[y]
                      + sR[wrapW(y + sw)] * maskv(y, -sw);
      num += I * ma;
      den += add * ma;
    }
    sFN[x] = clamp01(sF[x] * (num + EPS) / (den + EPS));
  }
  __syncthreads();

  // ---- stage 2: middle' (reads sM only at own x -> safe in place) ----
  for (int x = tid; x < Ww; x += 256) {
    float num = 0.f, den = 0.f;
#pragma unroll
    for (int i = 0; i < 2; ++i) {
      const int sw = SW[i];
      const float I = (i == 0) ? sI0[x] : sI1[x];
      const float add = sFN[wrapW(x - sw)] * maskv(x, sw) + sM[x]
                      + sR[wrapW(x + sw)] * maskv(x, -sw);
      num += I;
      den += add;
    }
    sM[x] = clamp01(sM[x] * (num + EPS) / (den + EPS));
  }
  __syncthreads();

  // ---- stage 3: rear' (uses new front sFN, new middle sM) -> sF ----
  for (int x = tid; x < Ww; x += 256) {
    float num = 0.f, den = 0.f;
#pragma unroll
    for (int i = 0; i < 2; ++i) {
      const int sw = SW[i];
      const int y = wrapW(x - sw);              // roll by sw
      const float mv = maskv(x, sw);
      const float I = (i == 0) ? sI0[y] : sI1[y];
      const float add = sFN[wrapW(y - sw)] * maskv(y, sw) + sM[y]
                      + sR[wrapW(y + sw)] * maskv(y, -sw);
      num += I * mv;
      den += add * mv;
    }
    sF[x] = clamp01(sR[x] * (num + EPS) / (den + EPS));
  }
  __syncthreads();  // also drains DScnt before async stores read LDS

  // ---- stage 4: async DMA LDS -> out, layout [layer, c, H, W] ----
  float* o0 = out + 0 * 3 * plane + rowoff;  // front'
  float* o1 = out + 1 * 3 * plane + rowoff;  // middle'
  float* o2 = out + 2 * 3 * plane + rowoff;  // rear'
  for (int q = tid; q < Ww / 4; q += 256) {
    const int x = q * 4;
    async_st_b128(o0 + x, lds_off(&sFN[x]));
    async_st_b128(o1 + x, lds_off(&sM[x]));
    async_st_b128(o2 + x, lds_off(&sF[x]));
  }
  wait_async0();
}

extern "C" void kernel_launch(void* const* d_in, const int* in_sizes, int n_in,
                              void* d_out, int out_size, void* d_ws, size_t ws_size,
                              hipStream_t stream) {
  const float* front  = (const float*)d_in[0];
  const float* middle = (const float*)d_in[1];
  const float* rear   = (const float*)d_in[2];
  const float* images = (const float*)d_in[3];
  float* out = (float*)d_out;

  ntf_row_kernel<<<dim3(3 * Hh), dim3(256), 0, stream>>>(
      front, middle, rear, images, out);
}